// Attention_Layer_34565896798209
// MI455X (gfx1250) — compile-verified
//
#include <hip/hip_runtime.h>
#include <hip/hip_bf16.h>
#include <math.h>

typedef __attribute__((ext_vector_type(2))) float v2f;
typedef __attribute__((ext_vector_type(8))) float v8f;

#define WMMA_F32X4(a, b, c) \
  __builtin_amdgcn_wmma_f32_16x16x4_f32(false, (a), false, (b), (short)0, (c), false, false)

static __device__ __forceinline__ v2f lds_v2(const float* p) {
  return *reinterpret_cast<const v2f*>(p);
}

constexpr int NB  = 8192;  // batches
constexpr int NA  = 64;    // agents
constexpr int DIN = 128;   // input dim
constexpr int E   = 32;    // embedding dim

// even pads -> 8B-aligned K-pair fragments, conflict-free strided access
constexpr int PW = DIN + 2;  // 130
constexpr int PE = E + 2;    // 34
constexpr int PN = NA + 2;   // 66

__global__ __launch_bounds__(128)
void attn_layer_kernel(const float* __restrict__ X,   // (NB*NA, DIN)
                       const float* __restrict__ W1,  // (DIN, E)
                       const float* __restrict__ b1,  // (E)
                       const float* __restrict__ Cm,  // (E, E)
                       const float* __restrict__ W2,  // (2E, 1)
                       const float* __restrict__ b2,  // (1)
                       float* __restrict__ out)       // (NB*NA)
{
  __shared__ __align__(16) float W1t[E][PW];    // W1 transposed: W1t[n][k]
  __shared__ __align__(16) float Ct[E][PE];     // C transposed:  Ct[n][k]
  __shared__ __align__(16) float fis[NA][PE];   // fi row-major   (A-frags + beta B-frags)
  __shared__ __align__(16) float fiT[E][PN];    // fi transposed  (v B-frags)
  __shared__ __align__(16) float gs[NA][PE];    // g row-major    (A-frags)
  __shared__ __align__(16) float als[NA][PN];   // alpha row-major(A-frags)
  __shared__ float W2s[2 * E];
  __shared__ float b1s[E];

  const int b    = blockIdx.x;
  const int tid  = threadIdx.x;
  const int lane = tid & 31;
  const int wave = tid >> 5;       // 0..3 -> 16-row strip
  const int half = lane >> 4;      // 0/1 (K pair select in A/B frags; M+8 in C/D)
  const int ln   = lane & 15;      // column / row-in-16 index
  const int m0   = wave * 16;
  const int kh   = half * 2;       // A/B frag: lanes 0-15 hold K0,K1; 16-31 hold K2,K3

  // ---- stage weights (transposed so B-frag K-pairs are contiguous) ----
  for (int i = tid; i < DIN * E; i += 128) W1t[i % E][i / E] = W1[i];
  for (int i = tid; i < E * E;   i += 128) Ct[i % E][i / E]  = Cm[i];
  if (tid < 2 * E) W2s[tid] = W2[tid];
  if (tid < E)     b1s[tid] = b1[tid];
  __syncthreads();

  // ---- fc1: fi = relu(X_b @ W1 + b1), strip rows m0..m0+15 ----
  const float* Xrow = X + ((size_t)b * NA + m0 + ln) * DIN;
  v8f acc0 = {}, acc1 = {};
  #pragma unroll 4
  for (int kk = 0; kk < DIN; kk += 4) {
    v2f a = *reinterpret_cast<const v2f*>(Xrow + kk + kh);
    v2f p = lds_v2(&W1t[ln][kk + kh]);
    v2f q = lds_v2(&W1t[16 + ln][kk + kh]);
    acc0 = WMMA_F32X4(a, p, acc0);
    acc1 = WMMA_F32X4(a, q, acc1);
  }
  #pragma unroll
  for (int i = 0; i < 8; ++i) {
    int m = m0 + half * 8 + i;           // C/D layout: VGPR i -> row i (+8 for upper half)
    float f0 = fmaxf(acc0[i] + b1s[ln],      0.f);
    float f1 = fmaxf(acc1[i] + b1s[16 + ln], 0.f);
    fis[m][ln]      = f0;  fis[m][16 + ln] = f1;
    fiT[ln][m]      = f0;  fiT[16 + ln][m] = f1;
  }
  __syncthreads();

  // ---- g = fi @ C (strip) ----
  v8f g0 = {}, g1 = {};
  #pragma unroll
  for (int kk = 0; kk < E; kk += 4) {
    v2f a = lds_v2(&fis[m0 + ln][kk + kh]);
    v2f p = lds_v2(&Ct[ln][kk + kh]);
    v2f q = lds_v2(&Ct[16 + ln][kk + kh]);
    g0 = WMMA_F32X4(a, p, g0);
    g1 = WMMA_F32X4(a, q, g1);
  }
  #pragma unroll
  for (int i = 0; i < 8; ++i) {
    int m = m0 + half * 8 + i;
    gs[m][ln] = g0[i];  gs[m][16 + ln] = g1[i];
  }
  __syncthreads();

  // ---- beta = g @ fi^T : strip rows x all 64 cols = 4 C-tiles ----
  v8f bt[4] = {{}, {}, {}, {}};
  #pragma unroll
  for (int kk = 0; kk < E; kk += 4) {
    v2f a = lds_v2(&gs[m0 + ln][kk + kh]);
    #pragma unroll
    for (int t = 0; t < 4; ++t) {
      v2f p = lds_v2(&fis[t * 16 + ln][kk + kh]);  // B[k][j] = fi[j][k]
      bt[t] = WMMA_F32X4(a, p, bt[t]);
    }
  }

  // ---- diagonal mask + softmax over 64 cols, in registers ----
  #pragma unroll
  for (int i = 0; i < 8; ++i) {
    const int m = m0 + half * 8 + i;     // agent index (row)
    #pragma unroll
    for (int t = 0; t < 4; ++t)
      if (t * 16 + ln == m) bt[t][i] = -INFINITY;
    float mx = fmaxf(fmaxf(bt[0][i], bt[1][i]), fmaxf(bt[2][i], bt[3][i]));
    #pragma unroll
    for (int s = 1; s < 16; s <<= 1) mx = fmaxf(mx, __shfl_xor(mx, s, 32));
    float sum = 0.f;
    #pragma unroll
    for (int t = 0; t < 4; ++t) { float e = __expf(bt[t][i] - mx); bt[t][i] = e; sum += e; }
    #pragma unroll
    for (int s = 1; s < 16; s <<= 1) sum += __shfl_xor(sum, s, 32);
    const float inv = 1.f / sum;
    #pragma unroll
    for (int t = 0; t < 4; ++t) { bt[t][i] *= inv; als[m][t * 16 + ln] = bt[t][i]; }
  }
  __syncthreads();

  // ---- v = alpha @ fi (strip) ----
  v8f v0 = {}, v1 = {};
  #pragma unroll
  for (int kk = 0; kk < NA; kk += 4) {
    v2f a = lds_v2(&als[m0 + ln][kk + kh]);
    v2f p = lds_v2(&fiT[ln][kk + kh]);        // B[k][e] = fi[k][e] via transposed copy
    v2f q = lds_v2(&fiT[16 + ln][kk + kh]);
    v0 = WMMA_F32X4(a, p, v0);
    v1 = WMMA_F32X4(a, q, v1);
  }

  // ---- out = sigmoid([fi, v] @ W2 + b2), 64-length dot per row ----
  const float b2v = b2[0];
  #pragma unroll
  for (int i = 0; i < 8; ++i) {
    const int m = m0 + half * 8 + i;
    float pp = fis[m][ln]      * W2s[ln]
             + fis[m][16 + ln] * W2s[16 + ln]
             + v0[i]           * W2s[E + ln]
             + v1[i]           * W2s[E + 16 + ln];
    #pragma unroll
    for (int s = 1; s < 16; s <<= 1) pp += __shfl_xor(pp, s, 32);
    if (ln == 0)
      out[(size_t)b * NA + m] = 1.f / (1.f + __expf(-(pp + b2v)));
  }
}

extern "C" void kernel_launch(void* const* d_in, const int* in_sizes, int n_in,
                              void* d_out, int out_size, void* d_ws, size_t ws_size,
                              hipStream_t stream) {
  const float* X  = (const float*)d_in[0];
  const float* W1 = (const float*)d_in[1];
  const float* b1 = (const float*)d_in[2];
  const float* Cm = (const float*)d_in[3];
  const float* W2 = (const float*)d_in[4];
  const float* b2 = (const float*)d_in[5];
  float* out = (float*)d_out;
  attn_layer_kernel<<<NB, 128, 0, stream>>>(X, W1, b1, Cm, W2, b2, out);
}